// LwLRAPLoss_36137854829035
// MI455X (gfx1250) — compile-verified
//
#include <hip/hip_runtime.h>
#include <stdint.h>

#define CC 512
#define TPB 256

__global__ void lwlrap_init(double* num, unsigned* den) {
  *num = 0.0;
  *den = 0u;
}

__global__ __launch_bounds__(TPB) void lwlrap_row_kernel(
    const float* __restrict__ preds, const float* __restrict__ labels,
    double* __restrict__ num_acc, unsigned* __restrict__ den_acc) {
  __shared__ float sp[CC];        // staged preds row
  __shared__ float sl[CC];        // staged labels row
  __shared__ uint64_t keys[CC];   // sortable keys
  __shared__ unsigned chcnt[16];  // per-32-chunk label counts
  __shared__ float wsum[8];       // per-wave partial sums

  const unsigned tid  = threadIdx.x;
  const unsigned row  = blockIdx.x;
  const unsigned lane = tid & 31u;
  const unsigned wv   = tid >> 5;

  // ---- Stage the row into LDS: one async B128 global->LDS load per thread ----
  // threads 0..127 stage preds (2KB), threads 128..255 stage labels (2KB).
  {
    const unsigned t2 = tid & 127u;
    const float* gsrc = (tid < 128u) ? (preds + (size_t)row * CC)
                                     : (labels + (size_t)row * CC);
    const uint64_t ga = (uint64_t)(uintptr_t)gsrc + 16ull * t2;
    float* ldst = (tid < 128u) ? &sp[4u * t2] : &sl[4u * t2];
    const uint32_t la = (uint32_t)(uintptr_t)ldst;
    asm volatile("global_load_async_to_lds_b128 %0, %1, off th:TH_LOAD_NT"
                 ::"v"(la), "v"(ga) : "memory");
    asm volatile("s_wait_asynccnt 0" ::: "memory");
  }
  __syncthreads();

  // ---- Build 64-bit keys: ascending sort => preds descending, ties by class idx ----
#pragma unroll
  for (int e = 0; e < 2; ++e) {
    unsigned i = tid + e * TPB;
    uint32_t fb = __float_as_uint(sp[i]);
    uint32_t s  = (fb & 0x80000000u) ? ~fb : (fb | 0x80000000u);  // monotonic asc
    uint32_t d  = ~s;                                             // descending order
    unsigned lab = (sl[i] != 0.0f) ? 1u : 0u;
    keys[i] = ((uint64_t)d << 32) | (uint64_t)((i << 1) | lab);
  }

  // ---- Bitonic sort (ascending) of 512 keys, 256 threads x 2 elements ----
  for (unsigned k = 2; k <= CC; k <<= 1) {
    for (unsigned j = k >> 1; j > 0; j >>= 1) {
      __syncthreads();
#pragma unroll
      for (int e = 0; e < 2; ++e) {
        unsigned i = tid + e * TPB;
        unsigned ixj = i ^ j;
        if (ixj > i) {
          uint64_t a = keys[i], b = keys[ixj];
          bool up = ((i & k) == 0);
          if ((a > b) == up) { keys[i] = b; keys[ixj] = a; }
        }
      }
    }
  }
  __syncthreads();

  // ---- Wave32 ballot scan over sorted label bits; accumulate cnt/rank ----
  unsigned b0 = (unsigned)(keys[tid] & 1ull);
  unsigned b1 = (unsigned)(keys[tid + TPB] & 1ull);
  unsigned m0 = (unsigned)__ballot(b0);  // wave32: low 32 bits
  unsigned m1 = (unsigned)__ballot(b1);
  if (lane == 0) {
    chcnt[wv]     = (unsigned)__popc(m0);
    chcnt[wv + 8] = (unsigned)__popc(m1);
  }
  __syncthreads();

  unsigned base0 = 0, base1 = 0;
#pragma unroll
  for (unsigned c = 0; c < 16; ++c) {
    unsigned v = chcnt[c];
    if (c < wv)     base0 += v;
    if (c < wv + 8) base1 += v;
  }
  unsigned pre0 = (unsigned)__popc(m0 & ((1u << lane) - 1u));
  unsigned pre1 = (unsigned)__popc(m1 & ((1u << lane) - 1u));

  float part = 0.0f;
  if (b0) part += (float)(base0 + pre0 + 1u) / (float)(tid + 1u);
  if (b1) part += (float)(base1 + pre1 + 1u) / (float)(tid + TPB + 1u);

  // ---- Wave32 butterfly reduction, then 8-way LDS combine ----
#pragma unroll
  for (int off = 16; off > 0; off >>= 1) part += __shfl_xor(part, off, 32);
  if (lane == 0) wsum[wv] = part;
  __syncthreads();

  if (tid == 0) {
    float bsum = 0.0f;
#pragma unroll
    for (int w = 0; w < 8; ++w) bsum += wsum[w];
    atomicAdd(num_acc, (double)bsum);
    unsigned tot = 0;
#pragma unroll
    for (unsigned c = 0; c < 16; ++c) tot += chcnt[c];
    atomicAdd(den_acc, tot);
  }
}

__global__ void lwlrap_finalize(const double* __restrict__ num,
                                const unsigned* __restrict__ den,
                                float* __restrict__ out) {
  unsigned d = *den;
  out[0] = (float)(*num / (double)(d ? d : 1u));
}

extern "C" void kernel_launch(void* const* d_in, const int* in_sizes, int n_in,
                              void* d_out, int out_size, void* d_ws, size_t ws_size,
                              hipStream_t stream) {
  const float* preds  = (const float*)d_in[0];
  const float* labels = (const float*)d_in[1];
  float* out = (float*)d_out;
  double* num   = (double*)d_ws;
  unsigned* den = (unsigned*)((char*)d_ws + sizeof(double));
  const int B = in_sizes[0] / CC;

  hipLaunchKernelGGL(lwlrap_init, dim3(1), dim3(1), 0, stream, num, den);
  hipLaunchKernelGGL(lwlrap_row_kernel, dim3((unsigned)B), dim3(TPB), 0, stream,
                     preds, labels, num, den);
  hipLaunchKernelGGL(lwlrap_finalize, dim3(1), dim3(1), 0, stream, num, den, out);
}